// MultipolarInteraction_7559142441140
// MI455X (gfx1250) — compile-verified
//
#include <hip/hip_runtime.h>
#include <math.h>

typedef __attribute__((ext_vector_type(2))) float v2f;
typedef __attribute__((ext_vector_type(8))) float v8f;

#define TPB 256

struct LL2 { long long x, y; };

// ---------------------------------------------------------------------------
// Kernel 1: stage coords into float4 (stride 4) and pack multipoles into
// stride-12 (16B-aligned) AoS so gathers become b128 loads.
// ---------------------------------------------------------------------------
__global__ void stage_kernel(const float* __restrict__ coords,
                             const float* __restrict__ q,
                             const float* __restrict__ p,
                             const float* __restrict__ t,
                             float4* __restrict__ c4,
                             float* __restrict__ mp, int natoms) {
    int a = blockIdx.x * blockDim.x + threadIdx.x;
    if (a >= natoms) return;

    float4 c;
    c.x = coords[3 * a + 0];
    c.y = coords[3 * a + 1];
    c.z = coords[3 * a + 2];
    c.w = 0.0f;
    c4[a] = c;

    const float* ta = t + (size_t)a * 9;
    const float third = 1.0f / 3.0f;
    float4 m0, m1, m2;
    m0.x = q[a];
    m0.y = p[3 * a + 0];
    m0.z = p[3 * a + 1];
    m0.w = p[3 * a + 2];
    m1.x = ta[0] * third;                 // xx
    m1.y = (ta[1] + ta[3]) * third;       // xy
    m1.z = (ta[2] + ta[6]) * third;       // xz
    m1.w = ta[4] * third;                 // yy
    m2.x = (ta[5] + ta[7]) * third;       // yz
    m2.y = ta[8] * third;                 // zz
    m2.z = 0.0f;
    m2.w = 0.0f;
    float4* o = (float4*)(mp + (size_t)a * 12);
    o[0] = m0; o[1] = m1; o[2] = m2;
}

// ---------------------------------------------------------------------------
// Kernel 2: per-pair energy + fp32-WMMA wave reduction
// ---------------------------------------------------------------------------
__global__ void pair_energy_kernel(const float4* __restrict__ c4,
                                   const float* __restrict__ box,
                                   const LL2* __restrict__ pairs,
                                   const float* __restrict__ mp,
                                   float* __restrict__ partials, int npairs) {
    int idx = blockIdx.x * blockDim.x + threadIdx.x;
    float e = 0.0f;

    if (idx < npairs) {
        LL2 pr = pairs[idx];                 // one b128 load
        int ai = (int)pr.x;
        int aj = (int)pr.y;
        float Lx = box[0], Ly = box[4], Lz = box[8];   // diagonal box
        float4 ci = c4[ai];                  // one b128 gather per side
        float4 cj = c4[aj];
        float x = cj.x - ci.x;
        float y = cj.y - ci.y;
        float z = cj.z - ci.z;
        // minimum image (RNE rounding matches jnp.round)
        x -= rintf(x * (1.0f / Lx)) * Lx;
        y -= rintf(y * (1.0f / Ly)) * Ly;
        z -= rintf(z * (1.0f / Lz)) * Lz;
        float r2 = x * x + y * y + z * z;

        if (r2 <= 100.0f && r2 > 0.0f) {     // dr <= CUTOFF; ~0.4% of pairs
            float dr    = sqrtf(r2);
            float drInv = 1.0f / dr;

            // Ewald damping (alpha = 0.3)
            float u  = 0.3f * dr;
            float u2 = u * u;
            float ex = expf(-u2);
            float ec = erfcf(u);
            float u3 = u2 * u, u5 = u3 * u2, u7 = u5 * u2;
            float g  = 1.1283791670955126f * ex;   // 2/sqrt(pi)*exp(-u^2)
            float dm1 = ec + g * u;
            float dm2 = ec + g * (u + 0.6666666666666667f * u3);
            float dm3 = ec + g * (u + 0.6666666666666667f * u3 + 0.26666666666666666f * u5);
            float dm4 = ec + g * (u + 0.6666666666666667f * u3 + 0.26666666666666666f * u5
                                    + 0.0761904761904762f * u7);

            float i2 = drInv * drInv;
            float i3 = i2 * drInv;
            float i5 = i3 * i2;
            float i7 = i5 * i2;
            float i9 = i7 * i2;
            float d0 = drInv * ec;
            float d3 = i3 * dm1;
            float d5 = i5 * dm2;
            float d7 = i7 * dm3;
            float d9 = i9 * dm4;

            float x2 = x * x, y2 = y * y, z2 = z * z;
            float xy = x * y, xz = x * z, yz = y * z;

            float tx = -x * d3, ty = -y * d3, tz = -z * d3;
            float txx = 3.0f * x2 * d5 - d3;
            float txy = 3.0f * xy * d5;
            float txz = 3.0f * xz * d5;
            float tyy = 3.0f * y2 * d5 - d3;
            float tyz = 3.0f * yz * d5;
            float tzz = 3.0f * z2 * d5 - d3;
            float txxx = -15.0f * x2 * x * d7 + 9.0f * x * d5;
            float txxy = -15.0f * x2 * y * d7 + 3.0f * y * d5;
            float txxz = -15.0f * x2 * z * d7 + 3.0f * z * d5;
            float tyyy = -15.0f * y2 * y * d7 + 9.0f * y * d5;
            float tyyx = -15.0f * y2 * x * d7 + 3.0f * x * d5;
            float tyyz = -15.0f * y2 * z * d7 + 3.0f * z * d5;
            float tzzz = -15.0f * z2 * z * d7 + 9.0f * z * d5;
            float tzzx = -15.0f * z2 * x * d7 + 3.0f * x * d5;
            float tzzy = -15.0f * z2 * y * d7 + 3.0f * y * d5;
            float txyz = -15.0f * x * y * z * d7;
            float txxxx = 105.0f * x2 * x2 * d9 - 90.0f * x2 * d7 + 9.0f * d5;
            float txxxy = 105.0f * x2 * xy * d9 - 45.0f * xy * d7;
            float txxxz = 105.0f * x2 * xz * d9 - 45.0f * xz * d7;
            float txxyy = 105.0f * x2 * y2 * d9 - 15.0f * (x2 + y2) * d7 + 3.0f * d5;
            float txxzz = 105.0f * x2 * z2 * d9 - 15.0f * (x2 + z2) * d7 + 3.0f * d5;
            float txxyz = 105.0f * x2 * yz * d9 - 15.0f * yz * d7;
            float tyyyy = 105.0f * y2 * y2 * d9 - 90.0f * y2 * d7 + 9.0f * d5;
            float tyyyx = 105.0f * y2 * xy * d9 - 45.0f * xy * d7;
            float tyyyz = 105.0f * y2 * yz * d9 - 45.0f * yz * d7;
            float tyyzz = 105.0f * y2 * z2 * d9 - 15.0f * (y2 + z2) * d7 + 3.0f * d5;
            float tyyxz = 105.0f * y2 * xz * d9 - 15.0f * xz * d7;
            float tzzzz = 105.0f * z2 * z2 * d9 - 90.0f * z2 * d7 + 9.0f * d5;
            float tzzzx = 105.0f * z2 * xz * d9 - 45.0f * xz * d7;
            float tzzzy = 105.0f * z2 * yz * d9 - 45.0f * yz * d7;
            float tzzxy = 105.0f * z2 * xy * d9 - 15.0f * xy * d7;

            // vectorized multipole gathers (3 x b128 per side)
            const float4* mi4 = (const float4*)(mp + (size_t)ai * 12);
            const float4* mj4 = (const float4*)(mp + (size_t)aj * 12);
            float4 ia = mi4[0], ib = mi4[1], icc = mi4[2];
            float4 ja = mj4[0], jb = mj4[1], jcc = mj4[2];
            float i0 = ia.x, i1v = ia.y, i2v = ia.z, i3v = ia.w;
            float i4 = ib.x, i5v = ib.y, i6 = ib.z, i7v = ib.w;
            float i8 = icc.x, i9v = icc.y;

            float r;
            r = d0*i0 - tx*i1v - ty*i2v - tz*i3v + txx*i4 + txy*i5v + txz*i6 + tyy*i7v + tyz*i8 + tzz*i9v;
            e += ja.x * r;
            r = tx*i0 - txx*i1v - txy*i2v - txz*i3v + txxx*i4 + txxy*i5v + txxz*i6 + tyyx*i7v + txyz*i8 + tzzx*i9v;
            e += ja.y * r;
            r = ty*i0 - txy*i1v - tyy*i2v - tyz*i3v + txxy*i4 + tyyx*i5v + txyz*i6 + tyyy*i7v + tyyz*i8 + tzzy*i9v;
            e += ja.z * r;
            r = tz*i0 - txz*i1v - tyz*i2v - tzz*i3v + txxz*i4 + txyz*i5v + tzzx*i6 + tyyz*i7v + tzzy*i8 + tzzz*i9v;
            e += ja.w * r;
            r = txx*i0 - txxx*i1v - txxy*i2v - txxz*i3v + txxxx*i4 + txxxy*i5v + txxxz*i6 + txxyy*i7v + txxyz*i8 + txxzz*i9v;
            e += jb.x * r;
            r = txy*i0 - txxy*i1v - tyyx*i2v - txyz*i3v + txxxy*i4 + txxyy*i5v + txxyz*i6 + tyyyx*i7v + tyyxz*i8 + tzzxy*i9v;
            e += jb.y * r;
            r = txz*i0 - txxz*i1v - txyz*i2v - tzzx*i3v + txxxz*i4 + txxyz*i5v + txxzz*i6 + tyyxz*i7v + tzzxy*i8 + tzzzx*i9v;
            e += jb.z * r;
            r = tyy*i0 - tyyx*i1v - tyyy*i2v - tyyz*i3v + txxyy*i4 + tyyyx*i5v + tyyxz*i6 + tyyyy*i7v + tyyyz*i8 + tyyzz*i9v;
            e += jb.w * r;
            r = tyz*i0 - txyz*i1v - tyyz*i2v - tzzy*i3v + txxyz*i4 + tyyxz*i5v + tzzxy*i6 + tyyyz*i7v + tyyzz*i8 + tzzzy*i9v;
            e += jcc.x * r;
            r = tzz*i0 - tzzx*i1v - tzzy*i2v - tzzz*i3v + txxzz*i4 + tzzxy*i5v + tzzzx*i6 + tyyzz*i7v + tzzzy*i8 + tzzzz*i9v;
            e += jcc.y * r;
        }
    }

    // ---- wave32 reduction via fp32 WMMA: D = A(16x4) * ones(4x16) ----------
    // A layout: lanes 0-15 supply A[L,0],A[L,1]; lanes 16-31 supply A[L-16,2],A[L-16,3].
    // e in slot 0, 0 in slot 1 -> row m sum = e(lane m) + e(lane m+16), exact fp32.
    v2f a; a[0] = e;    a[1] = 0.0f;
    v2f b; b[0] = 1.0f; b[1] = 1.0f;
    v8f c = {};
    c = __builtin_amdgcn_wmma_f32_16x16x4_f32(false, a, false, b, (short)0, c,
                                              false, false);
    float s = ((c[0] + c[1]) + (c[2] + c[3])) + ((c[4] + c[5]) + (c[6] + c[7]));
    s += __shfl_xor(s, 16, 32);           // fold the two row halves -> wave total

    __shared__ float wsum[TPB / 32];
    int lane = threadIdx.x & 31;
    int wid  = threadIdx.x >> 5;
    if (lane == 0) wsum[wid] = s;
    __syncthreads();
    if (threadIdx.x == 0) {
        float tsum = 0.0f;
        #pragma unroll
        for (int w = 0; w < TPB / 32; ++w) tsum += wsum[w];
        partials[blockIdx.x] = tsum;
    }
}

// ---------------------------------------------------------------------------
// Kernel 3: deterministic final reduction of per-block partials
// ---------------------------------------------------------------------------
__global__ void final_reduce_kernel(const float* __restrict__ partials, int n,
                                    float* __restrict__ out) {
    __shared__ float sh[TPB];
    float t = 0.0f;
    for (int i = threadIdx.x; i < n; i += TPB) t += partials[i];  // fixed order
    sh[threadIdx.x] = t;
    __syncthreads();
    for (int o = TPB / 2; o > 0; o >>= 1) {
        if (threadIdx.x < o) sh[threadIdx.x] += sh[threadIdx.x + o];
        __syncthreads();
    }
    if (threadIdx.x == 0) out[0] = 1.0f * sh[0];   // PREFACTOR = 1.0
}

// ---------------------------------------------------------------------------
extern "C" void kernel_launch(void* const* d_in, const int* in_sizes, int n_in,
                              void* d_out, int out_size, void* d_ws, size_t ws_size,
                              hipStream_t stream) {
    const float* coords = (const float*)d_in[0];
    const float* box    = (const float*)d_in[1];
    const LL2*   pairs  = (const LL2*)d_in[2];
    const float* q      = (const float*)d_in[3];
    const float* p      = (const float*)d_in[4];
    const float* t      = (const float*)d_in[5];

    int natoms = in_sizes[3];
    int npairs = in_sizes[2] / 2;

    // ws layout (floats): [coords4: natoms*4][mp12: natoms*12][partials]
    float4* c4       = (float4*)d_ws;
    float*  mp       = (float*)d_ws + (size_t)natoms * 4;
    float*  partials = (float*)d_ws + (size_t)natoms * 16;

    int stageBlocks = (natoms + TPB - 1) / TPB;
    int nblocks     = (npairs + TPB - 1) / TPB;

    stage_kernel<<<stageBlocks, TPB, 0, stream>>>(coords, q, p, t, c4, mp, natoms);
    pair_energy_kernel<<<nblocks, TPB, 0, stream>>>(c4, box, pairs, mp,
                                                    partials, npairs);
    final_reduce_kernel<<<1, TPB, 0, stream>>>(partials, nblocks, (float*)d_out);
}